// MDGCL_26706106646803
// MI455X (gfx1250) — compile-verified
//
#include <hip/hip_runtime.h>
#include <hip/hip_bf16.h>
#include <math.h>

// ---------------- problem constants (match reference) ----------------
#define N_U   100000
#define N_I   50000
#define DD    64
#define DS    65          // D+1
#define NE    2000000
#define NL    2
#define NB    2048
#define NK    10
#define P_DROP 0.2f
#define INV_KEEP 1.25f    // 1/(1-P_DROP)
#define INV_T  5.0f       // 1/0.2 (both TEMP and T_GRAPH)
#define LHYPER 0.1f
#define LGRAPH 1.0f
#define LDK   68          // padded row stride for similarity operands (8B aligned)
#define KSTEPS 17         // 68/4 k-steps per WMMA chain

typedef __attribute__((ext_vector_type(2))) float v2f;
typedef __attribute__((ext_vector_type(8))) float v8f;

// ---------------------------------------------------------------------
// Edge-parallel SpMM with hardware f32 atomics. One wave per edge.
// stride==64: float2 per lane (b64 loads, 2 atomics). stride==65: strided.
// Dropped edges (val==0) retire the whole wave early (uniform branch).
// ---------------------------------------------------------------------
__global__ __launch_bounds__(256)
void spmm_atomic_kernel(float* __restrict__ out, const float* __restrict__ X,
                        const int* __restrict__ seg, const int* __restrict__ gat,
                        const float* __restrict__ vals, const float* __restrict__ kr,
                        int nE, int stride)
{
    int e    = (blockIdx.x * blockDim.x + threadIdx.x) >> 5;
    int lane = threadIdx.x & 31;
    if (e >= nE) return;
    float v = vals[e];
    if (kr) v = (kr[e] >= P_DROP) ? v * INV_KEEP : 0.0f;
    if (v == 0.0f) return;
    int s = seg[e], g = gat[e];
    if (stride == 64) {
        const float2 x = *(const float2*)(X + (size_t)g * 64 + lane * 2);
        float* o = out + (size_t)s * 64 + lane * 2;
        unsafeAtomicAdd(o,     v * x.x);
        unsafeAtomicAdd(o + 1, v * x.y);
    } else {
        for (int c = lane; c < 65; c += 32)
            unsafeAtomicAdd(out + (size_t)s * 65 + c, v * X[(size_t)g * 65 + c]);
    }
}

// z = act(tmp) + Zc ; Zc += z ; Zl = z ; Eacc += z ; Etot += Eacc
__global__ __launch_bounds__(256)
void layer_update_kernel(const float* __restrict__ tmp, float* __restrict__ Zc,
                         float* __restrict__ Zl, float* __restrict__ Eacc,
                         float* __restrict__ Etot, int n)
{
    int i = blockIdx.x * blockDim.x + threadIdx.x;
    if (i >= n) return;
    float t = tmp[i];
    float z = (t >= 0.0f ? t : 0.5f * t) + Zc[i];
    Zc[i] += z;
    Zl[i] = z;
    float e2 = Eacc[i] + z;
    Eacc[i] = e2;
    if (Etot) Etot[i] += e2;
}

// gather+normalize two 64-wide rows; write stride-68 zero-padded; diag = <ga,gb>
__global__ __launch_bounds__(64)
void prep_graph_kernel(const float* __restrict__ Za, const float* __restrict__ Zb,
                       const int* __restrict__ ids, float* __restrict__ Ga,
                       float* __restrict__ Gb, float* __restrict__ diag)
{
    int b = blockIdx.x, t = threadIdx.x;
    __shared__ float sa[64], sb[64], red[64];
    __shared__ float na, nb;
    int id = ids[b];
    float va = Za[(size_t)id * 64 + t], vb = Zb[(size_t)id * 64 + t];
    sa[t] = va; sb[t] = vb;
    __syncthreads();
    if (t == 0) {
        float s0 = 0.f, s1 = 0.f;
        for (int i = 0; i < 64; ++i) { s0 += sa[i] * sa[i]; s1 += sb[i] * sb[i]; }
        na = fmaxf(sqrtf(s0), 1e-12f); nb = fmaxf(sqrtf(s1), 1e-12f);
    }
    __syncthreads();
    float ga = va / na, gb = vb / nb;
    Ga[(size_t)b * LDK + t] = ga; Gb[(size_t)b * LDK + t] = gb;
    if (t < 4) { Ga[(size_t)b * LDK + 64 + t] = 0.f; Gb[(size_t)b * LDK + 64 + t] = 0.f; }
    red[t] = ga * gb;
    __syncthreads();
    if (t == 0) { float d = 0.f; for (int i = 0; i < 64; ++i) d += red[i]; diag[b] = d; }
}

// hyperbolic prep: g = normalize([0,z]); h = normalize(expmap0(sz)) @ W ; diag = <g,h>
__global__ __launch_bounds__(128)
void prep_hyper_kernel(const float* __restrict__ Z, const float* __restrict__ Sz,
                       const float* __restrict__ W, const int* __restrict__ ids,
                       float* __restrict__ G, float* __restrict__ H,
                       float* __restrict__ diag)
{
    int b = blockIdx.x, t = threadIdx.x;
    __shared__ float z[64], s[65], e[65], g[65], h[65];
    __shared__ float zn;
    int id = ids[b];
    if (t < 64) z[t] = Z[(size_t)id * 64 + t];
    if (t < 65) s[t] = Sz[(size_t)id * 65 + t];
    __syncthreads();
    if (t == 0) {
        float q = 0.f; for (int i = 0; i < 64; ++i) q += z[i] * z[i];
        zn = fmaxf(sqrtf(q), 1e-12f);
        float x2 = 0.f; for (int i = 1; i < 65; ++i) x2 += s[i] * s[i];
        float xn = fmaxf(sqrtf(x2), 1e-7f);
        float sh = sinhf(xn) / xn;
        e[0] = coshf(xn);
        for (int i = 1; i < 65; ++i) e[i] = sh * s[i];
        float en = 0.f; for (int i = 0; i < 65; ++i) en += e[i] * e[i];
        en = fmaxf(sqrtf(en), 1e-12f);
        for (int i = 0; i < 65; ++i) e[i] /= en;
    }
    __syncthreads();
    if (t < 65) { float gv = (t == 0) ? 0.0f : z[t - 1] / zn; g[t] = gv; G[(size_t)b * LDK + t] = gv; }
    if (t >= 65 && t < LDK) G[(size_t)b * LDK + t] = 0.0f;
    __syncthreads();
    if (t < 65) {
        float acc = 0.f;
        for (int i = 0; i < 65; ++i) acc += e[i] * W[i * 65 + t];
        h[t] = acc; H[(size_t)b * LDK + t] = acc;
    }
    if (t >= 65 && t < LDK) H[(size_t)b * LDK + t] = 0.0f;
    __syncthreads();
    if (t == 0) { float d = 0.f; for (int i = 0; i < 65; ++i) d += g[i] * h[i]; diag[b] = d; }
}

// ---------------------------------------------------------------------
// Fused 2048x2048xK f32 similarity + streaming exp row-sum via WMMA.
// One wave per 16-row strip. A frag preloaded (17 v2f regs), B frags
// streamed per 16-col tile. C layout: VGPR v -> row v+8*(lane>>4),
// col = lane&15; width-16 shfl_xor reduces per-row partial exp sums.
// mode 0: += log(rowsum) - diag/T   (graph logsumexp, diag-shift folded)
// mode 1: += -log(exp(diag/T)/(rowsum+1e-8)+1e-8)  (hyperbolic InfoNCE)
// |args| <= 10 so exp is safe in f32 without max-subtraction.
// ---------------------------------------------------------------------
__global__ __launch_bounds__(32)
void sim_lse_wmma_kernel(const float* __restrict__ A, const float* __restrict__ Bm,
                         const float* __restrict__ diag, float invT, int mode,
                         float* __restrict__ outslot)
{
    const int lane = threadIdx.x;
    const int r  = lane & 15;
    const int hi = lane >> 4;
    const int rtile = blockIdx.x;

    const float* arow = A + (size_t)(rtile * 16 + r) * LDK + hi * 2;
    v2f a[KSTEPS];
#pragma unroll
    for (int sct = 0; sct < KSTEPS; ++sct)
        a[sct] = *(const v2f*)(arow + sct * 4);

    float racc[8] = {0.f, 0.f, 0.f, 0.f, 0.f, 0.f, 0.f, 0.f};

    for (int ct = 0; ct < NB / 16; ++ct) {
        const float* brow = Bm + (size_t)(ct * 16 + r) * LDK + hi * 2;
        __builtin_prefetch(brow + 16 * LDK, 0, 1);
        v8f c = {0.f, 0.f, 0.f, 0.f, 0.f, 0.f, 0.f, 0.f};
#pragma unroll
        for (int sct = 0; sct < KSTEPS; ++sct) {
            v2f bv = *(const v2f*)(brow + sct * 4);
            c = __builtin_amdgcn_wmma_f32_16x16x4_f32(
                    false, a[sct], false, bv, (short)0, c, false, false);
        }
#pragma unroll
        for (int v = 0; v < 8; ++v)
            racc[v] += __expf(c[v] * invT);
    }
#pragma unroll
    for (int v = 0; v < 8; ++v) {
        racc[v] += __shfl_xor(racc[v], 8, 16);
        racc[v] += __shfl_xor(racc[v], 4, 16);
        racc[v] += __shfl_xor(racc[v], 2, 16);
        racc[v] += __shfl_xor(racc[v], 1, 16);
    }
    if (r == 0) {
        float loc = 0.f;
#pragma unroll
        for (int v = 0; v < 8; ++v) {
            int row = rtile * 16 + hi * 8 + v;
            float sum = racc[v];
            float d   = diag[row];
            if (mode == 0) loc += logf(sum) - d * invT;
            else           loc += -logf(__expf(d * invT) / (sum + 1e-8f) + 1e-8f);
        }
        unsafeAtomicAdd(outslot, loc);
    }
}

// BPR ranking loss: sum relu(1 - <u,p> + <u,n>) / B  -> out[1]
__global__ __launch_bounds__(64)
void loss_r_kernel(const float* __restrict__ EuT, const float* __restrict__ EiT,
                   const int* __restrict__ uids, const int* __restrict__ pos,
                   const int* __restrict__ neg, float* __restrict__ out)
{
    int b = blockIdx.x, t = threadIdx.x;
    __shared__ float u[64], red[64];
    u[t] = EuT[(size_t)uids[b] * 64 + t];
    __syncthreads();
    float acc = 0.f;
    for (int k = 0; k < NK; ++k) {
        int p = pos[b * NK + k], n = neg[b * NK + k];
        red[t] = u[t] * (EiT[(size_t)n * 64 + t] - EiT[(size_t)p * 64 + t]);
        __syncthreads();
        for (int off = 32; off > 0; off >>= 1) {
            if (t < off) red[t] += red[t + off];
            __syncthreads();
        }
        if (t == 0) acc += fmaxf(0.0f, 1.0f + red[0]);
        __syncthreads();
    }
    if (t == 0) unsafeAtomicAdd(&out[1], acc / (float)NB);
}

__global__ void combine_kernel(float* out)
{
    if (threadIdx.x == 0) out[0] = out[1] + LHYPER * out[2] + LGRAPH * out[3];
}

// ---------------------------------------------------------------------
extern "C" void kernel_launch(void* const* d_in, const int* in_sizes, int n_in,
                              void* d_out, int out_size, void* d_ws, size_t ws_size,
                              hipStream_t stream)
{
    const float* E_u_0 = (const float*)d_in[0];
    const float* E_i_0 = (const float*)d_in[1];
    const float* S_u_0 = (const float*)d_in[2];
    const float* S_i_0 = (const float*)d_in[3];
    const float* Ws    = (const float*)d_in[4];
    const float* adjv  = (const float*)d_in[5];
    const float* krand = (const float*)d_in[6];
    const int*   rows  = (const int*)d_in[7];
    const int*   cols  = (const int*)d_in[8];
    const int*   uids  = (const int*)d_in[9];
    const int*   iids  = (const int*)d_in[10];
    const int*   pos   = (const int*)d_in[11];
    const int*   neg   = (const int*)d_in[12];
    float* out = (float*)d_out;

    const size_t UD = (size_t)N_U * 64, ID = (size_t)N_I * 64;
    const size_t US = (size_t)N_U * 65, IS = (size_t)N_I * 65;

    float* cur = (float*)d_ws;
    auto alloc = [&](size_t n) { float* p = cur; cur += n; return p; };
    float *Eu = alloc(UD), *Euu = alloc(UD), *EuT = alloc(UD), *Zu_c = alloc(UD), *Zuu_c = alloc(UD);
    float *Ei = alloc(ID), *Eii = alloc(ID), *EiT = alloc(ID), *Zi_c = alloc(ID), *Zii_c = alloc(ID);
    float *Su = alloc(US), *Szu_c = alloc(US);
    float *Si = alloc(IS), *Szi_c = alloc(IS);
    float *ZuL[NL], *ZiL[NL], *SzuL[NL], *SziL[NL], *ZuuL[NL], *ZiiL[NL];
    for (int k = 0; k < NL; ++k) {
        ZuL[k] = alloc(UD);  ZuuL[k] = alloc(UD); SzuL[k] = alloc(US);
        ZiL[k] = alloc(ID);  ZiiL[k] = alloc(ID); SziL[k] = alloc(IS);
    }
    float *t_zu = alloc(UD), *t_zuu = alloc(UD), *t_zus = alloc(US);
    float *t_zi = alloc(ID), *t_zii = alloc(ID), *t_zis = alloc(IS);
    float *GA = alloc((size_t)NB * LDK), *GB = alloc((size_t)NB * LDK), *dg = alloc(NB);

    // init state
    hipMemsetAsync(out, 0, 4 * sizeof(float), stream);
    hipMemcpyAsync(Eu,  E_u_0, UD * 4, hipMemcpyDeviceToDevice, stream);
    hipMemcpyAsync(Euu, E_u_0, UD * 4, hipMemcpyDeviceToDevice, stream);
    hipMemcpyAsync(EuT, E_u_0, UD * 4, hipMemcpyDeviceToDevice, stream);
    hipMemcpyAsync(Ei,  E_i_0, ID * 4, hipMemcpyDeviceToDevice, stream);
    hipMemcpyAsync(Eii, E_i_0, ID * 4, hipMemcpyDeviceToDevice, stream);
    hipMemcpyAsync(EiT, E_i_0, ID * 4, hipMemcpyDeviceToDevice, stream);
    hipMemcpyAsync(Su, S_u_0, US * 4, hipMemcpyDeviceToDevice, stream);
    hipMemcpyAsync(Si, S_i_0, IS * 4, hipMemcpyDeviceToDevice, stream);
    hipMemsetAsync(Zu_c, 0, UD * 4, stream);  hipMemsetAsync(Zuu_c, 0, UD * 4, stream);
    hipMemsetAsync(Zi_c, 0, ID * 4, stream);  hipMemsetAsync(Zii_c, 0, ID * 4, stream);
    hipMemsetAsync(Szu_c, 0, US * 4, stream); hipMemsetAsync(Szi_c, 0, IS * 4, stream);

    const int eBlocks = (NE + 7) / 8;   // 8 edge-waves per 256-thread block
    auto nb256 = [](size_t n) { return (int)((n + 255) / 256); };

    for (int layer = 0; layer < NL; ++layer) {
        const float* kr0 = krand + (size_t)(4 * layer + 0) * NE;
        const float* kr1 = krand + (size_t)(4 * layer + 1) * NE;
        const float* kr2 = krand + (size_t)(4 * layer + 2) * NE;
        const float* kr3 = krand + (size_t)(4 * layer + 3) * NE;
        hipMemsetAsync(t_zu, 0, UD * 4, stream);  hipMemsetAsync(t_zuu, 0, UD * 4, stream);
        hipMemsetAsync(t_zi, 0, ID * 4, stream);  hipMemsetAsync(t_zii, 0, ID * 4, stream);
        hipMemsetAsync(t_zus, 0, US * 4, stream); hipMemsetAsync(t_zis, 0, IS * 4, stream);

        spmm_atomic_kernel<<<eBlocks, 256, 0, stream>>>(t_zu,  Ei,  rows, cols, adjv, kr0, NE, 64);
        spmm_atomic_kernel<<<eBlocks, 256, 0, stream>>>(t_zi,  Eu,  cols, rows, adjv, kr1, NE, 64);
        spmm_atomic_kernel<<<eBlocks, 256, 0, stream>>>(t_zus, Si,  rows, cols, adjv, nullptr, NE, 65);
        spmm_atomic_kernel<<<eBlocks, 256, 0, stream>>>(t_zis, Su,  cols, rows, adjv, nullptr, NE, 65);
        spmm_atomic_kernel<<<eBlocks, 256, 0, stream>>>(t_zuu, Eii, rows, cols, adjv, kr2, NE, 64);
        spmm_atomic_kernel<<<eBlocks, 256, 0, stream>>>(t_zii, Euu, cols, rows, adjv, kr3, NE, 64);

        layer_update_kernel<<<nb256(UD), 256, 0, stream>>>(t_zu,  Zu_c,  ZuL[layer],  Eu,  EuT,    (int)UD);
        layer_update_kernel<<<nb256(ID), 256, 0, stream>>>(t_zi,  Zi_c,  ZiL[layer],  Ei,  EiT,    (int)ID);
        layer_update_kernel<<<nb256(US), 256, 0, stream>>>(t_zus, Szu_c, SzuL[layer], Su,  nullptr,(int)US);
        layer_update_kernel<<<nb256(IS), 256, 0, stream>>>(t_zis, Szi_c, SziL[layer], Si,  nullptr,(int)IS);
        layer_update_kernel<<<nb256(UD), 256, 0, stream>>>(t_zuu, Zuu_c, ZuuL[layer], Euu, nullptr,(int)UD);
        layer_update_kernel<<<nb256(ID), 256, 0, stream>>>(t_zii, Zii_c, ZiiL[layer], Eii, nullptr,(int)ID);
    }

    loss_r_kernel<<<NB, 64, 0, stream>>>(EuT, EiT, uids, pos, neg, out);

    for (int k = 0; k < NL; ++k) {
        // graph-contrastive (mode 0) -> out[3]
        prep_graph_kernel<<<NB, 64, 0, stream>>>(ZuL[k], ZuuL[k], uids, GA, GB, dg);
        sim_lse_wmma_kernel<<<NB / 16, 32, 0, stream>>>(GA, GB, dg, INV_T, 0, &out[3]);
        prep_graph_kernel<<<NB, 64, 0, stream>>>(ZiL[k], ZiiL[k], iids, GA, GB, dg);
        sim_lse_wmma_kernel<<<NB / 16, 32, 0, stream>>>(GA, GB, dg, INV_T, 0, &out[3]);
        // hyperbolic (mode 1) -> out[2]
        const float* Wk = Ws + (size_t)k * 65 * 65;
        prep_hyper_kernel<<<NB, 128, 0, stream>>>(ZuL[k], SzuL[k], Wk, uids, GA, GB, dg);
        sim_lse_wmma_kernel<<<NB / 16, 32, 0, stream>>>(GA, GB, dg, INV_T, 1, &out[2]);
        prep_hyper_kernel<<<NB, 128, 0, stream>>>(ZiL[k], SziL[k], Wk, iids, GA, GB, dg);
        sim_lse_wmma_kernel<<<NB / 16, 32, 0, stream>>>(GA, GB, dg, INV_T, 1, &out[2]);
    }

    combine_kernel<<<1, 32, 0, stream>>>(out);
    (void)in_sizes; (void)n_in; (void)out_size; (void)ws_size;
}